// RCNNMultiBinDetLabelFromMatch_15719580304265
// MI455X (gfx1250) — compile-verified
//
#include <hip/hip_runtime.h>
#include <hip/hip_bf16.h>
#include <math.h>

// RCNN multi-bin detection label generation for MI455X (gfx1250).
//
// Bandwidth analysis: outputs ~202 MB vs ~0.4 MB inputs -> pure streaming-store
// kernel, HBM floor ~= 202MB / 23.3TB/s ~= 8.7us. Strategy:
//   * separable gaussian -> rank-1 outer product via V_WMMA_F32_16X16X4_F32
//     (one WMMA per anchor produces the whole 16x16 map in the C/D register
//     layout; lane = column, vgpr row + lane-half = row)
//   * __builtin_nontemporal_store for every output (no reuse; output ~= L2 size)
//   * one wave32 per anchor, no divergence (branchless exp/selects, rcp not div)

typedef float v2f __attribute__((ext_vector_type(2)));
typedef float v8f __attribute__((ext_vector_type(8)));

#define FWH 16
#define NC 8
#define ZOOM_W 1.1f
#define ZOOM_H 1.1f

__global__ __launch_bounds__(256) void rcnn_label_wmma_kernel(
    const float* __restrict__ boxes,     // (B,N,4)
    const float* __restrict__ gt_boxes,  // (B,G,5)
    const int*   __restrict__ flag,      // (B,N)
    const int*   __restrict__ gid,       // (B,N)
    float* __restrict__ out_label,       // (B*N, 8, 16, 16)
    float* __restrict__ out_off,         // (B*N, 4, 16, 16)
    float* __restrict__ out_mask,        // (B*N, 8)
    int N, int G)
{
    const int tid  = threadIdx.x;
    const int wave = tid >> 5;
    const int lane = tid & 31;
    const int a    = blockIdx.x * 8 + wave;     // anchor index in [0, B*N)
    const int b    = a / N;

    // ---- per-anchor scalar inputs (uniform across the wave) ----
    const float4 bx  = reinterpret_cast<const float4*>(boxes)[a];
    const int    gix = gid[a];
    const int    fl  = flag[a];
    const float* gt  = gt_boxes + (size_t)(b * G + gix) * 5;
    const float gx1 = gt[0], gy1 = gt[1], gx2 = gt[2], gy2 = gt[3], label = gt[4];

    // ---- per-anchor scalar math (redundant per lane; trivially cheap) ----
    const float cx = (bx.x + bx.z) * 0.5f, cy = (bx.y + bx.w) * 0.5f;
    const float hw = (bx.z - bx.x) * (ZOOM_W * 0.5f);
    const float hh = (bx.w - bx.y) * (ZOOM_H * 0.5f);
    const float ax1 = cx - hw, ay1 = cy - hh, ax2 = cx + hw, ay2 = cy + hh;

    const float rx1 = gx1 - ax1, ry1 = gy1 - ay1;
    const float rx2 = gx2 - ax1, ry2 = gy2 - ay1;
    const float rw  = rx2 - rx1, rh  = ry2 - ry1;
    const float rcx = (rx1 + rx2) * 0.5f, rcy = (ry1 + ry2) * 0.5f;
    const float sw  = (ax2 - ax1) * (1.0f / FWH);
    const float sh  = (ay2 - ay1) * (1.0f / FWH);

    // v_rcp_f32 instead of IEEE-division macro-sequences (no branches, ~1 ulp)
    const float inv_sw = __builtin_amdgcn_rcpf(sw);
    const float inv_sh = __builtin_amdgcn_rcpf(sh);

    const float sig_w = rw * 0.5f * inv_sw, sig_h = rh * 0.5f * inv_sh;
    const float inv_sig_w = __builtin_amdgcn_rcpf(sig_w);
    const float inv_sig_h = __builtin_amdgcn_rcpf(sig_h);
    const float pw = rcx * inv_sw, ph = rcy * inv_sh;
    const bool  small = (sw < 0.01f) || (sh < 0.01f);
    const float fcls  = fabsf(label) - 1.0f;   // float class index in [0,7]

    // ---- separable gaussian: lane computes column/row factor for i=lane&15 ----
    // Branchless: always evaluate v_exp_f32, then v_cndmask on the window test.
    const int   col = lane & 15;
    const float fc  = (float)col + 0.5f;
    const float uw  = (pw - fc) * inv_sig_w;
    const float ew  = __expf(-(uw * uw));
    const float gw_l = (fabsf(fc - pw) < sig_w) ? ew : 0.0f;
    const float uh  = (ph - fc) * inv_sig_h;
    const float eh  = __expf(-(uh * uh));
    const float gh_l = (fabsf(fc - ph) < sig_h) ? eh : 0.0f;

    // ---- rank-1 outer product on the matrix unit ----
    // Replicate gh into every A slot and gw/4 into every B slot so that
    // A[m][k] = gh[m] for all k and B[k][n] = gw[n]/4 for all k, making
    // D[m][n] = sum_k gh[m]*gw[n]/4 = gh[m]*gw[n] independent of the exact
    // K <-> {vgpr, lane-half} mapping. Lane gives n, vgpr row + half gives m.
    v2f Amat; Amat.x = gh_l;          Amat.y = gh_l;
    v2f Bmat; Bmat.x = gw_l * 0.25f;  Bmat.y = gw_l * 0.25f;
    v8f Cacc = {};
    v8f D = __builtin_amdgcn_wmma_f32_16x16x4_f32(
        /*neg_a=*/false, Amat, /*neg_b=*/false, Bmat,
        /*c_mod=*/(short)0, Cacc, /*reuse_a=*/false, /*reuse_b=*/false);

    // ---- per-class selectors (loop-invariant across rows) ----
    bool mc[NC];
#pragma unroll
    for (int c = 0; c < NC; ++c) mc[c] = ((float)c == fcls);

    // ---- streaming stores (non-temporal: output has zero reuse) ----
    float* lm = out_label + (size_t)a * (NC * 256);
    float* of = out_off   + (size_t)a * (4  * 256);
    const int half8 = (lane >> 4) << 3;               // row block: 0 or 8

    const float zmask = small ? 0.0f : 1.0f;          // branchless small-gate
    const float o0 = zmask * (rx1 * inv_sw - fc);     // depends on column only
    const float o2 = zmask * (rx2 * inv_sw - fc);
    const float y1s = ry1 * inv_sh, y2s = ry2 * inv_sh;

#pragma unroll
    for (int r = 0; r < 8; ++r) {
        const int   m    = r + half8;                 // feature-map row
        const int   idx  = m * 16 + col;
        const float g    = zmask * D[r];
        const float rowc = (float)m + 0.5f;
        const float o1   = zmask * (y1s - rowc);
        const float o3   = zmask * (y2s - rowc);

#pragma unroll
        for (int c = 0; c < NC; ++c) {
            const float v = mc[c] ? g : 0.0f;
            __builtin_nontemporal_store(v, lm + c * 256 + idx);
        }
        __builtin_nontemporal_store(o0, of + 0 * 256 + idx);
        __builtin_nontemporal_store(o1, of + 1 * 256 + idx);
        __builtin_nontemporal_store(o2, of + 2 * 256 + idx);
        __builtin_nontemporal_store(o3, of + 3 * 256 + idx);
    }

    // ---- mask: (flag>0) & (label>0) & (class match); NOT gated by `small` ----
    if (lane < NC) {
        const bool mm = (fl > 0) && (label > 0.0f) && mc[lane];
        __builtin_nontemporal_store(mm ? 1.0f : 0.0f, out_mask + (size_t)a * NC + lane);
    }
}

extern "C" void kernel_launch(void* const* d_in, const int* in_sizes, int n_in,
                              void* d_out, int out_size, void* d_ws, size_t ws_size,
                              hipStream_t stream) {
    const float* boxes    = (const float*)d_in[0];   // (B,N,4) f32
    const float* gt_boxes = (const float*)d_in[1];   // (B,G,5) f32
    const int*   flag     = (const int*)d_in[2];     // (B,N)   i32
    const int*   gid      = (const int*)d_in[3];     // (B,N)   i32

    const int B  = 16;
    const int BN = in_sizes[2];            // B*N = 16384
    const int N  = BN / B;                 // 1024
    const int G  = in_sizes[1] / (B * 5);  // 64

    float* out       = (float*)d_out;
    float* out_label = out;                                   // BN*8*256
    float* out_off   = out_label + (size_t)BN * NC * 256;     // BN*4*256
    float* out_mask  = out_off   + (size_t)BN * 4  * 256;     // BN*8

    dim3 grid(BN / 8);   // 8 waves (anchors) per 256-thread block
    dim3 block(256);
    rcnn_label_wmma_kernel<<<grid, block, 0, stream>>>(
        boxes, gt_boxes, flag, gid, out_label, out_off, out_mask, N, G);
}